// DepthFusionNet_88012469830583
// MI455X (gfx1250) — compile-verified
//
#include <hip/hip_runtime.h>
#include <stdint.h>

// ---------------- problem constants (fixed by the reference module) ----------
constexpr int B  = 16;
constexpr int N  = 500000;
constexpr int H  = 352;
constexpr int W  = 1216;
constexpr int DH = (int)(0.2 / 2 * H);   // 35
constexpr int DW = (int)(0.2 / 2 * W);   // 121
constexpr int HF = H + 2 * DH;           // 422
constexpr int WF = W + 2 * DW;           // 1458
constexpr float INV_MAX_DEPTH = 1.0f / 50.0f;

// ---------------- tiling ----------------------------------------------------
constexpr int TILE   = 256;                        // points per wave-tile (TDM DMA = 3 KB)
constexpr int WAVES  = 8;                          // 256-thread block = 8 wave32
constexpr int CHUNKS = (N + TILE - 1) / TILE;      // 1954 chunks per batch
constexpr int TOTAL_TILES = B * CHUNKS;            // 31264
constexpr int GRID = TOTAL_TILES / (2 * WAVES);    // 1954 blocks -> exactly 2 tiles/wave

typedef unsigned int u32x4 __attribute__((ext_vector_type(4)));
typedef int          i32x4 __attribute__((ext_vector_type(4)));
typedef int          i32x8 __attribute__((ext_vector_type(8)));

#if __has_builtin(__builtin_amdgcn_tensor_load_to_lds) && __has_builtin(__builtin_amdgcn_s_wait_tensorcnt)
#define USE_TDM 1
#else
#define USE_TDM 0
#endif

#if USE_TDM
// Issue one TDM descriptor: 2D tile (TILE points x 3 channels) of the pcd
// tensor for `tile` = b*CHUNKS + chunk, into LDS at byte offset lds_byte_off.
// tensor_dim0 = remaining points -> ragged tail reads return 0 (z=0 => dropped).
__device__ __forceinline__
void tdm_issue_tile(const float* pcd, unsigned lds_byte_off, int tile) {
  const int b     = tile / CHUNKS;
  const int chunk = tile - b * CHUNKS;
  const uint64_t elem  = (uint64_t)(unsigned)(b * 3 * N) + (unsigned)(chunk * TILE);
  const uint64_t gaddr = (uint64_t)pcd + 4ull * elem;       // tile start (x-channel)
  const unsigned rem   = (unsigned)(N - chunk * TILE);      // valid extent in dim0

  // D# group 0: [1:0]=count=1 | [63:32]=lds_addr | [120:64]=global_addr | [127:126]=type=2
  u32x4 g0 = { 1u,
               lds_byte_off,
               (unsigned)gaddr,
               (((unsigned)(gaddr >> 32)) & 0x01FFFFFFu) | 0x80000000u };

  // D# group 1:
  //  [17:16] data_size=2 (4B)   [79:48] tensor_dim0=rem   [111:80] tensor_dim1=3
  //  [127:112] tile_dim0=TILE   [143:128] tile_dim1=3     [207:160] dim0_stride=N
  i32x8 g1 = { (int)0x00020000u,
               (int)((rem & 0xFFFFu) << 16),
               (int)((rem >> 16) | (3u << 16)),
               (int)((unsigned)TILE << 16),
               3,
               N,
               0, 0 };

  i32x4 z4 = { 0, 0, 0, 0 };
#if defined(__clang_major__) && (__clang_major__ >= 23)
  i32x8 z8 = { 0, 0, 0, 0, 0, 0, 0, 0 };
  __builtin_amdgcn_tensor_load_to_lds(g0, g1, z4, z4, z8, 0);
#else
  __builtin_amdgcn_tensor_load_to_lds(g0, g1, z4, z4, 0);
#endif
}
#endif

// ---------------- project + scatter (emitted first -> visible in disasm) -----
__global__ __launch_bounds__(WAVES * 32, 1) void depth_scatter_kernel(
    const float* __restrict__ pcd,
    const float* __restrict__ fx, const float* __restrict__ fy,
    const float* __restrict__ cx, const float* __restrict__ cy,
    float* __restrict__ out)
{
#if USE_TDM
  __shared__ float lds[WAVES * 2 * 3 * TILE];   // 48 KB: 2 buffers x 3 KB per wave

  const int lane = (int)(threadIdx.x & 31u);
  const int wave = __builtin_amdgcn_readfirstlane((int)threadIdx.x) >> 5;
  const int wbase = wave * (2 * 3 * TILE);      // wave's region (float index)

  // Flat LDS addresses carry the LDS byte offset in the low 32 bits (aperture map).
  const unsigned wavebuf_byte = (unsigned)(uintptr_t)(void*)&lds[wbase];

  // Escape the LDS region so the compiler cannot assume it is never written
  // (the TDM writes it); reads stay direct shared-array indexing -> ds_load_b32.
  { float* p = &lds[wbase]; asm volatile("" : "+v"(p)); }

  int t = (int)blockIdx.x * WAVES + wave;
  const int stride = (int)gridDim.x * WAVES;
  if (t >= TOTAL_TILES) return;

  tdm_issue_tile(pcd, wavebuf_byte, t);          // prologue: fill buffer 0
  int bi = 0;

  for (;;) {
    const int  nt        = t + stride;
    const bool have_next = (nt < TOTAL_TILES);
    if (have_next) {
      tdm_issue_tile(pcd, wavebuf_byte + (unsigned)((bi ^ 1) * 3 * TILE * 4), nt);
      __builtin_amdgcn_s_wait_tensorcnt(1);      // oldest (current tile) DMA done
    } else {
      __builtin_amdgcn_s_wait_tensorcnt(0);
    }
    asm volatile("" ::: "memory");               // reloads below the wait, each iter

    const int   b   = t / CHUNKS;
    const float fxb = fx[b], fyb = fy[b], cxb = cx[b], cyb = cy[b];

    const int base = wbase + bi * 3 * TILE;      // TDM packed rows: x | y | z

    #pragma unroll
    for (int i = 0; i < TILE / 32; ++i) {
      const int   k = lane + 32 * i;             // bank-conflict-free stride
      const float x = lds[base + k];
      const float y = lds[base + TILE + k];
      const float z = lds[base + 2 * TILE + k];
      const float iz = __builtin_amdgcn_rcpf(z); // 1-ulp rcp; exactness irrelevant pre-cast
      const int px = (int)(fxb * x * iz + cxb);  // trunc-toward-zero == astype(int32)
      const int py = (int)(fyb * y * iz + cyb);
      if ((z > 0.0f) & (px >= -DW) & (px < W + DW) & (py >= -DH) & (py < H + DH)) {
        out[(b * HF + (py + DH)) * WF + (px + DW)] = z * INV_MAX_DEPTH;  // L2-resident scatter
      }
    }

    if (!have_next) break;
    t = nt; bi ^= 1;
  }
#else
  // Fallback: direct coalesced global loads (no TDM builtin on this toolchain).
  int i = (int)(blockIdx.x * blockDim.x + threadIdx.x);
  const int stride = (int)(gridDim.x * blockDim.x);
  for (; i < B * N; i += stride) {
    const int b = i / N;
    const int n = i - b * N;
    const float x = pcd[(b * 3 + 0) * N + n];
    const float y = pcd[(b * 3 + 1) * N + n];
    const float z = pcd[(b * 3 + 2) * N + n];
    const float iz = __builtin_amdgcn_rcpf(z);
    const int px = (int)(fx[b] * x * iz + cx[b]);
    const int py = (int)(fy[b] * y * iz + cy[b]);
    if ((z > 0.0f) & (px >= -DW) & (px < W + DW) & (py >= -DH) & (py < H + DH)) {
      out[(b * HF + (py + DH)) * WF + (px + DW)] = z * INV_MAX_DEPTH;
    }
  }
#endif
}

// ---------------- zero the output image (39.4 MB, b128 stores) ---------------
__global__ void zero_f32_kernel(float* __restrict__ out, int n) {
  int i = (blockIdx.x * blockDim.x + threadIdx.x) * 4;
  if (i + 4 <= n) {
    float4 z; z.x = 0.f; z.y = 0.f; z.z = 0.f; z.w = 0.f;
    *reinterpret_cast<float4*>(out + i) = z;
  } else {
    for (; i < n; ++i) out[i] = 0.0f;
  }
}

// ---------------- launcher ---------------------------------------------------
extern "C" void kernel_launch(void* const* d_in, const int* in_sizes, int n_in,
                              void* d_out, int out_size, void* d_ws, size_t ws_size,
                              hipStream_t stream) {
  (void)in_sizes; (void)n_in; (void)d_ws; (void)ws_size;

  const float* pcd = (const float*)d_in[0];
  const float* fx  = (const float*)d_in[1];
  const float* fy  = (const float*)d_in[2];
  const float* cx  = (const float*)d_in[3];
  const float* cy  = (const float*)d_in[4];
  float* out = (float*)d_out;

  // 1) zero the image (required every call; harness does not re-poison).
  const int nq = (out_size + 3) / 4;
  zero_f32_kernel<<<dim3((nq + 255) / 256), dim3(256), 0, stream>>>(out, out_size);

  // 2) TDM-staged project + scatter (runs after zeroing; same stream).
  depth_scatter_kernel<<<dim3(GRID), dim3(WAVES * 32), 0, stream>>>(pcd, fx, fy, cx, cy, out);
}